// delta_q_params_41162966564948
// MI455X (gfx1250) — compile-verified
//
#include <hip/hip_runtime.h>
#include <cmath>

// Problem dims (fixed by the reference)
#define T_DIM   16384
#define D_DIM   256
#define H_DIM   512
#define G_DIM   1536     // 3H
#define ZD      64
#define OUT_DIM 2144     // Z + Z(Z+1)/2
#define C_DIM   1024     // 2H
#define NWG_DIR 16       // workgroups per direction in the scan

typedef float v2f __attribute__((ext_vector_type(2)));
typedef float v8f __attribute__((ext_vector_type(8)));
typedef unsigned int v4u __attribute__((ext_vector_type(4)));
typedef int v4i __attribute__((ext_vector_type(4)));
typedef int v8i __attribute__((ext_vector_type(8)));

#if defined(__gfx1250__) && __has_builtin(__builtin_amdgcn_wmma_f32_16x16x4_f32)
#define HAVE_WMMA_F32X4 1
#endif

#if defined(__gfx1250__) && __has_builtin(__builtin_amdgcn_tensor_load_to_lds)
#define HAVE_TDM 1
#else
#define HAVE_TDM 0
#endif

// ---------------------------------------------------------------------------
// f32 WMMA 16x16x4 (frag layouts per CDNA5 ISA 7.12.2)
// ---------------------------------------------------------------------------
__device__ __forceinline__ v8f wmma4(v2f a, v2f b, v8f c) {
#ifdef HAVE_WMMA_F32X4
  return __builtin_amdgcn_wmma_f32_16x16x4_f32(false, a, false, b, (short)0, c,
                                               false, false);
#else
  const int lane = __lane_id();
  const int n = lane & 15;
  const int half = lane >> 4;
  #pragma unroll
  for (int i = 0; i < 8; ++i) {
    const int m = half * 8 + i;
    #pragma unroll
    for (int k = 0; k < 4; ++k) {
      const int grp = (k >> 1) * 16;
      float av = __shfl((k & 1) ? a.y : a.x, m + grp, 32);
      float bv = __shfl((k & 1) ? b.y : b.x, n + grp, 32);
      c[i] += av * bv;
    }
  }
  return c;
#endif
}

// ---------------------------------------------------------------------------
// Tensor Data Mover: 2D tile (tile_d0 contiguous elems x tile_d1 rows of f32,
// row stride = stride0 elems) global -> LDS, optional HW padding in LDS
// (pad_interval_code: 4 -> pad 1 dword every 32, 5 -> every 64).
// D# layout per CDNA5 ISA 8.3/8.4.
// ---------------------------------------------------------------------------
#if HAVE_TDM
__device__ __forceinline__ void tdm_load_2d(unsigned int lds_off,
                                            const void* gaddr,
                                            unsigned int tile_d0,
                                            unsigned int tile_d1,
                                            unsigned long long stride0,
                                            unsigned int pad_code,
                                            unsigned int pad_en) {
  unsigned long long ga = (unsigned long long)(uintptr_t)gaddr;
  v4u g0;
  g0[0] = 1u;                                           // count=1 (user D#)
  g0[1] = lds_off;                                      // lds_addr (bytes)
  g0[2] = (unsigned int)(ga & 0xffffffffu);             // global_addr[31:0]
  g0[3] = (unsigned int)((ga >> 32) & 0x01ffffffu) | (2u << 30);  // type=2
  v8i g1;
  g1[0] = (int)((2u << 16) | (pad_en << 20) | ((pad_code & 7u) << 22));
  g1[1] = (int)((tile_d0 & 0xffffu) << 16);             // tensor_dim0 = tile_d0
  g1[2] = (int)((tile_d1 & 0xffffu) << 16);             // tensor_dim1 = tile_d1
  g1[3] = (int)((tile_d0 & 0xffffu) << 16);             // tile_dim0
  g1[4] = (int)(tile_d1 & 0xffffu);                     // tile_dim1 (tile_dim2=0)
  g1[5] = (int)(unsigned int)(stride0 & 0xffffffffull); // dim0_stride[31:0]
  g1[6] = (int)(unsigned int)((stride0 >> 32) & 0xffffull);
  g1[7] = 0;
  v4i z4 = {};
#if __clang_major__ >= 23
  v8i z8 = {};
  __builtin_amdgcn_tensor_load_to_lds(g0, g1, z4, z4, z8, 0);
#else
  __builtin_amdgcn_tensor_load_to_lds(g0, g1, z4, z4, 0);
#endif
}
#endif

// ---------------------------------------------------------------------------
// Tiled f32 GEMM: C[M,N] = (revA ? reverse_rows(A) : A) * B (+bias)
// BM=128 x BN=32 macro-tile, K-chunk 64, TDM double-buffered LDS staging,
// 8 waves x 16x32 strip (2 accumulators share one A fragment).
// M%128==0, N%32==0, K%64==0 (holds for all GEMMs here).
// ---------------------------------------------------------------------------
#define BM 128
#define BN 32
#define BK 64
#define LDA_S 65
#define LDB_S 33

__global__ void __launch_bounds__(256)
gemm_wmma_f32(const float* __restrict__ A, const float* __restrict__ B,
              const float* __restrict__ bias, float* __restrict__ C,
              int M, int N, int K, int revA) {
  extern __shared__ float gsm[];
  float* AsBase = gsm;                        // 2 * [128][65]
  float* BsBase = gsm + 2 * BM * LDA_S;       // 2 * [64][33]
  const int tid  = threadIdx.x;
  const int bm0  = blockIdx.y * BM;
  const int bn0  = blockIdx.x * BN;
  const int w    = tid >> 5;
  const int lane = tid & 31;
  const int ln   = lane & 15;
  const int half = lane >> 4;
  const int tm   = w * 16;
  const int nchunks = K / BK;
  const size_t aGlobRow0 = revA ? (size_t)(M - BM - bm0) : (size_t)bm0;
  const int aRow    = tm + ln;
  const int aRowLds = revA ? (BM - 1 - aRow) : aRow;  // LDS rows ascend in memory
  v8f acc0 = {}, acc1 = {};

#if HAVE_TDM
  if (w == 0) {  // wave-uniform: only wave 0 drives the TDM pipeline
    tdm_load_2d((unsigned int)(uintptr_t)AsBase, A + aGlobRow0 * K,
                BK, BM, (unsigned long long)K, 5u, 1u);
    tdm_load_2d((unsigned int)(uintptr_t)BsBase, B + bn0,
                BN, BK, (unsigned long long)N, 4u, 1u);
  }
#endif
  for (int ci = 0; ci < nchunks; ++ci) {
    const int buf = ci & 1;
    float* As = AsBase + buf * BM * LDA_S;
    float* Bs = BsBase + buf * BK * LDB_S;
#if HAVE_TDM
    if (w == 0) {
      if (ci + 1 < nchunks) {
        const int nb = buf ^ 1;
        tdm_load_2d((unsigned int)(uintptr_t)(AsBase + nb * BM * LDA_S),
                    A + aGlobRow0 * K + (size_t)(ci + 1) * BK,
                    BK, BM, (unsigned long long)K, 5u, 1u);
        tdm_load_2d((unsigned int)(uintptr_t)(BsBase + nb * BK * LDB_S),
                    B + (size_t)(ci + 1) * BK * N + bn0,
                    BN, BK, (unsigned long long)N, 4u, 1u);
        __builtin_amdgcn_s_wait_tensorcnt(2);   // chunk ci resident
      } else {
        __builtin_amdgcn_s_wait_tensorcnt(0);
      }
    }
#else
    for (int idx = tid; idx < BM * BK; idx += 256) {
      int r = idx >> 6, c = idx & 63;
      As[r * LDA_S + c] = A[(aGlobRow0 + r) * K + (size_t)ci * BK + c];
    }
    for (int idx = tid; idx < BK * BN; idx += 256) {
      int r = idx >> 5, c = idx & 31;
      Bs[r * LDB_S + c] = B[(size_t)(ci * BK + r) * N + bn0 + c];
    }
#endif
    __syncthreads();
    const float* Ap = As + aRowLds * LDA_S;
    #pragma unroll
    for (int kk = 0; kk < 16; ++kk) {
      int k0 = kk * 4 + 2 * half;
      v2f a  = { Ap[k0], Ap[k0 + 1] };
      v2f b0 = { Bs[k0 * LDB_S + ln],      Bs[(k0 + 1) * LDB_S + ln] };
      v2f b1 = { Bs[k0 * LDB_S + 16 + ln], Bs[(k0 + 1) * LDB_S + 16 + ln] };
      acc0 = wmma4(a, b0, acc0);
      acc1 = wmma4(a, b1, acc1);
    }
    __syncthreads();
  }
  #pragma unroll
  for (int i = 0; i < 8; ++i) {
    int row = bm0 + tm + half * 8 + i;
    int c0 = bn0 + ln, c1 = bn0 + 16 + ln;
    float v0 = acc0[i], v1 = acc1[i];
    if (bias) { v0 += bias[c0]; v1 += bias[c1]; }
    C[(size_t)row * N + c0] = v0;
    C[(size_t)row * N + c1] = v1;
  }
}

// ---------------------------------------------------------------------------
// Bidirectional GRU scan. 32 blocks: dir = blk>>4, g = blk&15. Each block owns
// 32 hidden units; its 96 Wh columns land in LDS via 3 TDM tiles in
// [gate][k][ju] layout (lane-consecutive -> conflict-free). Per step: h_prev
// (2 KB, L2-resident) -> LDS, 96 threads do one 512-long dot each, 32 threads
// gate + write h_new, then a monotonic global-atomic phase barrier (s_sleep
// spin) syncs the direction. Next gi row is prefetched during compute.
// ---------------------------------------------------------------------------
__device__ __forceinline__ float sigmoidf_(float x) {
  return 1.0f / (1.0f + __expf(-x));
}

__global__ void __launch_bounds__(128)
gru_scan(const float* __restrict__ gi_f, const float* __restrict__ gi_b,
         const float* __restrict__ Wh_f, const float* __restrict__ bh_f,
         const float* __restrict__ Wh_b, const float* __restrict__ bh_b,
         float* __restrict__ hcat, unsigned int* __restrict__ ctr) {
  extern __shared__ float smem[];
  float* Whs = smem;                       // 3 * [512][32]
  float* hs  = Whs + 3 * 512 * 32;         // 512
  float* ghs = hs + 512;                   // 96
  float* bhs = ghs + 96;                   // 96

  const int tid = threadIdx.x;
  const int dir = blockIdx.x >> 4;         // 0 = fwd, 1 = bwd
  const int g   = blockIdx.x & (NWG_DIR - 1);
  const int u0  = g * 32;

  const float* gi = dir ? gi_b : gi_f;
  const float* Wh = dir ? Wh_b : Wh_f;
  const float* bh = dir ? bh_b : bh_f;

  // Stage Wh slice once: gate q tile = Wh[k][q*H + u0 + ju], 512 x 32
#if HAVE_TDM
  if (tid < 32) {
    for (int q = 0; q < 3; ++q)
      tdm_load_2d((unsigned int)(uintptr_t)(Whs + q * 512 * 32),
                  Wh + (size_t)q * H_DIM + u0, 32, 512,
                  (unsigned long long)G_DIM, 0u, 0u);
    __builtin_amdgcn_s_wait_tensorcnt(0);
  }
#else
  for (int idx = tid; idx < 3 * 512 * 32; idx += 128) {
    int q = idx >> 14, rem = idx & 16383, k = rem >> 5, ju = rem & 31;
    Whs[idx] = Wh[(size_t)k * G_DIM + q * H_DIM + u0 + ju];
  }
#endif
  if (tid < 96) bhs[tid] = bh[(tid >> 5) * H_DIM + u0 + (tid & 31)];
  __syncthreads();

  const int q = tid >> 5, ju = tid & 31;

  for (int s = 0; s < T_DIM; ++s) {
    // h_prev (fwd: row s-1 cols[0:512]; bwd: row T-s cols[512:1024])
    if (s == 0) {
      for (int i = tid; i < H_DIM; i += 128) hs[i] = 0.0f;
    } else {
      const float* hp = dir ? (hcat + (size_t)(T_DIM - s) * C_DIM + H_DIM)
                            : (hcat + (size_t)(s - 1) * C_DIM);
      for (int i = tid; i < H_DIM; i += 128) hs[i] = hp[i];
    }
    __syncthreads();

    const float* girow = gi + (size_t)s * G_DIM;
    if (tid >= 96 && s + 1 < T_DIM)   // prefetch next gi row behind compute
      __builtin_prefetch(girow + G_DIM + (tid - 96) * 48, 0, 3);

    if (tid < 96) {
      const float* wc = Whs + q * (512 * 32) + ju;
      float acc = bhs[tid];
      #pragma unroll 8
      for (int k = 0; k < H_DIM; ++k) acc = fmaf(wc[k * 32], hs[k], acc);
      ghs[tid] = acc;
    }
    __syncthreads();

    if (tid < 32) {
      int jj = u0 + tid;
      float r = sigmoidf_(girow[jj] + ghs[tid]);
      float z = sigmoidf_(girow[H_DIM + jj] + ghs[32 + tid]);
      float n = tanhf(girow[2 * H_DIM + jj] + r * ghs[64 + tid]);
      float hn = (1.0f - z) * n + z * hs[jj];
      float* ho = dir ? (hcat + (size_t)(T_DIM - 1 - s) * C_DIM + H_DIM + jj)
                      : (hcat + (size_t)s * C_DIM + jj);
      *ho = hn;
    }
    __threadfence();
    __syncthreads();

    if (s < T_DIM - 1) {
      if (tid == 0) {
        __hip_atomic_fetch_add(&ctr[dir], 1u, __ATOMIC_RELEASE,
                               __HIP_MEMORY_SCOPE_AGENT);
        unsigned target = (unsigned)(s + 1) * NWG_DIR;
        while (__hip_atomic_load(&ctr[dir], __ATOMIC_ACQUIRE,
                                 __HIP_MEMORY_SCOPE_AGENT) < target)
          __builtin_amdgcn_s_sleep(2);
      }
      __syncthreads();
    }
  }
}

__global__ void init_ctr(unsigned int* c) {
  if (threadIdx.x < 2) c[threadIdx.x] = 0u;
}

// ---------------------------------------------------------------------------
// Per-timestep factor kernel (one 256-thread block per t):
//   L from flat (softplus diag) -> Sigma = L*L^T (WMMA) -> Cholesky(Sigma+eps)
//   -> Linv -> J = Linv^T*Linv (WMMA) -> h = Linv^T (Linv mu)
// ---------------------------------------------------------------------------
__global__ void __launch_bounds__(256)
factor_kernel(const float* __restrict__ outbuf, float* __restrict__ Sig_o,
              float* __restrict__ mu_o, float* __restrict__ J_o,
              float* __restrict__ h_o) {
  __shared__ float Ls[ZD][ZD + 1];   // L, later reused as Linv
  __shared__ float Ss[ZD][ZD + 1];   // Sigma+eps, in-place Cholesky
  __shared__ float mus[ZD];
  __shared__ float vs[ZD];

  const int t   = blockIdx.x;
  const int tid = threadIdx.x;
  const float* row = outbuf + (size_t)t * OUT_DIM;

  const int w    = tid >> 5;
  const int lane = tid & 31;
  const int ln   = lane & 15;
  const int half = lane >> 4;

  for (int i = tid; i < ZD * (ZD + 1); i += 256) ((float*)Ls)[i] = 0.0f;
  __syncthreads();

  for (int p = tid; p < 2080; p += 256) {
    int r = (int)((sqrtf(8.0f * (float)p + 1.0f) - 1.0f) * 0.5f);
    while ((r + 1) * (r + 2) / 2 <= p) ++r;
    while (r * (r + 1) / 2 > p) --r;
    int c = p - r * (r + 1) / 2;
    Ls[r][c] = row[ZD + p];
  }
  if (tid < ZD) {
    mus[tid] = row[tid];
    mu_o[(size_t)t * ZD + tid] = row[tid];
  }
  __syncthreads();
  if (tid < ZD) {
    float d = Ls[tid][tid];
    Ls[tid][tid] = (d > 20.0f) ? d : log1pf(__expf(d));  // softplus
  }
  __syncthreads();

  // Sigma = L * L^T via WMMA: 16 tiles, 8 waves x 2
  for (int tt = 0; tt < 2; ++tt) {
    int tile = w + tt * 8;
    int tm = (tile >> 2) * 16, tn = (tile & 3) * 16;
    v8f acc = {};
    #pragma unroll
    for (int kk = 0; kk < 16; ++kk) {
      int k0 = kk * 4 + 2 * half;
      v2f a = { Ls[tm + ln][k0], Ls[tm + ln][k0 + 1] };
      v2f b = { Ls[tn + ln][k0], Ls[tn + ln][k0 + 1] };  // B[k][n] = L[n][k]
      acc = wmma4(a, b, acc);
    }
    #pragma unroll
    for (int i = 0; i < 8; ++i) {
      int ri = tm + half * 8 + i, ci = tn + ln;
      float v = acc[i];
      Sig_o[(size_t)t * (ZD * ZD) + ri * ZD + ci] = v;  // Sigma w/o boost
      Ss[ri][ci] = v + ((ri == ci) ? 1e-9f : 0.0f);     // boosted for chol
    }
  }
  __syncthreads();

  // In-place Cholesky (lower) on Ss
  for (int k = 0; k < ZD; ++k) {
    if (tid == 0) Ss[k][k] = sqrtf(Ss[k][k]);
    __syncthreads();
    float inv = 1.0f / Ss[k][k];
    for (int i = k + 1 + tid; i < ZD; i += 256) Ss[i][k] *= inv;
    __syncthreads();
    for (int j = k + 1; j < ZD; ++j)
      for (int i = j + tid; i < ZD; i += 256)
        Ss[i][j] -= Ss[i][k] * Ss[j][k];
    __syncthreads();
  }

  // Linv (lower) into Ls: column j solved by thread j
  for (int i = tid; i < ZD * (ZD + 1); i += 256) ((float*)Ls)[i] = 0.0f;
  __syncthreads();
  if (tid < ZD) {
    int j = tid;
    for (int i = j; i < ZD; ++i) {
      float s = (i == j) ? 1.0f : 0.0f;
      for (int k2 = j; k2 < i; ++k2) s -= Ss[i][k2] * Ls[k2][j];
      Ls[i][j] = s / Ss[i][i];
    }
  }
  __syncthreads();

  // J = Linv^T * Linv via WMMA
  for (int tt = 0; tt < 2; ++tt) {
    int tile = w + tt * 8;
    int tm = (tile >> 2) * 16, tn = (tile & 3) * 16;
    v8f acc = {};
    #pragma unroll
    for (int kk = 0; kk < 16; ++kk) {
      int k0 = kk * 4 + 2 * half;
      v2f a = { Ls[k0][tm + ln], Ls[k0 + 1][tm + ln] };  // A[i][k]=Linv[k][i]
      v2f b = { Ls[k0][tn + ln], Ls[k0 + 1][tn + ln] };  // B[k][n]=Linv[k][n]
      acc = wmma4(a, b, acc);
    }
    #pragma unroll
    for (int i = 0; i < 8; ++i) {
      int ri = tm + half * 8 + i, ci = tn + ln;
      J_o[(size_t)t * (ZD * ZD) + ri * ZD + ci] = acc[i];
    }
  }

  // h = Linv^T * (Linv * mu)
  if (tid < ZD) {
    float s = 0.0f;
    for (int j = 0; j <= tid; ++j) s = fmaf(Ls[tid][j], mus[j], s);
    vs[tid] = s;
  }
  __syncthreads();
  if (tid < ZD) {
    float s = 0.0f;
    for (int k = tid; k < ZD; ++k) s = fmaf(Ls[k][tid], vs[k], s);
    h_o[(size_t)t * ZD + tid] = s;
  }
}

// ---------------------------------------------------------------------------
// Host orchestration. Workspace (floats):
//   [0, T*G)          gi_f   (reused as `out` buffer after the scan)
//   [T*G, 2*T*G)      gi_b   (tail reused by `out` as well)
//   [2*T*G, +T*1024)  hcat = concat(h_fwd, h_bwd)
//   then 2 uints      scan phase counters          (peak ~268.5 MB)
// ---------------------------------------------------------------------------
extern "C" void kernel_launch(void* const* d_in, const int* in_sizes, int n_in,
                              void* d_out, int out_size, void* d_ws,
                              size_t ws_size, hipStream_t stream) {
  (void)in_sizes; (void)n_in; (void)out_size; (void)ws_size;
  const float* y    = (const float*)d_in[0];
  const float* Wi_f = (const float*)d_in[1];
  const float* Wh_f = (const float*)d_in[2];
  const float* bh_f = (const float*)d_in[3];
  const float* Wi_b = (const float*)d_in[4];
  const float* Wh_b = (const float*)d_in[5];
  const float* bh_b = (const float*)d_in[6];
  const float* Wd   = (const float*)d_in[7];
  const float* bd   = (const float*)d_in[8];

  float* ws   = (float*)d_ws;
  float* gi_f = ws;
  float* gi_b = ws + (size_t)T_DIM * G_DIM;
  float* hcat = ws + (size_t)2 * T_DIM * G_DIM;
  unsigned int* ctr = (unsigned int*)(hcat + (size_t)T_DIM * C_DIM);
  float* outbuf = gi_f;  // 2*T*G floats >= T*OUT_DIM, free after the scan

  float* o     = (float*)d_out;
  float* Sig_o = o;
  float* mu_o  = Sig_o + (size_t)T_DIM * ZD * ZD;
  float* J_o   = mu_o + (size_t)T_DIM * ZD;
  float* h_o   = J_o + (size_t)T_DIM * ZD * ZD;

  init_ctr<<<1, 32, 0, stream>>>(ctr);

  const size_t gemm_lds = (size_t)(2 * BM * LDA_S + 2 * BK * LDB_S) * sizeof(float);

  // gi = y @ Wi (backward direction uses reversed rows of y)
  gemm_wmma_f32<<<dim3(G_DIM / BN, T_DIM / BM), 256, gemm_lds, stream>>>(
      y, Wi_f, nullptr, gi_f, T_DIM, G_DIM, D_DIM, 0);
  gemm_wmma_f32<<<dim3(G_DIM / BN, T_DIM / BM), 256, gemm_lds, stream>>>(
      y, Wi_b, nullptr, gi_b, T_DIM, G_DIM, D_DIM, 1);

  // bidirectional recurrent scan (16 WGs/direction, Wh resident in LDS via TDM)
  const size_t scan_lds = (size_t)(3 * 512 * 32 + 512 + 96 + 96) * sizeof(float);
  gru_scan<<<dim3(2 * NWG_DIR), 128, scan_lds, stream>>>(
      gi_f, gi_b, Wh_f, bh_f, Wh_b, bh_b, hcat, ctr);

  // out = carry @ Wd + bd   (carry == hcat, K = 1024)
  gemm_wmma_f32<<<dim3(OUT_DIM / BN, T_DIM / BM), 256, gemm_lds, stream>>>(
      hcat, Wd, bd, outbuf, T_DIM, OUT_DIM, C_DIM, 0);

  // per-timestep Sigma / Cholesky / inverse / h
  factor_kernel<<<dim3(T_DIM), 256, 0, stream>>>(outbuf, Sig_o, mu_o, J_o, h_o);
}